// SemanticFusionModule_11235634446448
// MI455X (gfx1250) — compile-verified
//
#include <hip/hip_runtime.h>
#include <hip/hip_bf16.h>

#define NNODES 100000
#define NHE    20000
#define NE     800000
#define DD     128
#define CCHUNK 512

typedef float v2f __attribute__((ext_vector_type(2)));
typedef float v8f __attribute__((ext_vector_type(8)));

// ---- ordered-int encoding for float atomicMax ----
__device__ __forceinline__ unsigned encOrd(float f) {
  unsigned u = __float_as_uint(f);
  return (u & 0x80000000u) ? ~u : (u | 0x80000000u);
}
__device__ __forceinline__ float decOrd(unsigned k) {
  return (k & 0x80000000u) ? __uint_as_float(k & 0x7fffffffu)
                           : __uint_as_float(~k);
}

__global__ void fill_f32(float* p, int n, float v) {
  int i = blockIdx.x * blockDim.x + threadIdx.x;
  if (i < n) p[i] = v;
}
__global__ void fill_u32(unsigned* p, int n, unsigned v) {
  int i = blockIdx.x * blockDim.x + threadIdx.x;
  if (i < n) p[i] = v;
}

// column sums & sums of squares over [nrows, 128]
__global__ void colsum_kernel(const float* __restrict__ x, int nrows,
                              float* __restrict__ cs, float* __restrict__ csq) {
  int c = threadIdx.x;  // 128 threads per block, one column each
  float s = 0.f, sq = 0.f;
  for (int r = blockIdx.x; r < nrows; r += gridDim.x) {
    float v = x[(long)r * DD + c];
    s += v; sq += v * v;
  }
  atomicAdd(&cs[c], s);
  atomicAdd(&csq[c], sq);
}

// fold GraphNorm into the GEMM: gn(x) = a*x + c per column
// Wsc[k][n] = a_k * W[k][n];  cW[n] = sum_k c_k * W[k][n]
__global__ void prep_kernel(const float* __restrict__ cs, const float* __restrict__ csq,
                            const float* __restrict__ gw, const float* __restrict__ gb,
                            const float* __restrict__ gms, const float* __restrict__ W,
                            float* __restrict__ Wsc, float* __restrict__ cW, float invN) {
  int k = threadIdx.x;  // 128 threads
  float mean = cs[k] * invN;
  float ex2  = csq[k] * invN;
  float ms   = gms[k];
  // var = E[(x - ms*mean)^2] = E[x^2] - 2*ms*mean^2 + ms^2*mean^2
  float var = ex2 - 2.f * ms * mean * mean + ms * ms * mean * mean;
  float a = gw[k] * rsqrtf(var + 1e-5f);
  float c = gb[k] - a * ms * mean;
  __shared__ float csh[DD];
  csh[k] = c;
  for (int n = 0; n < DD; ++n) Wsc[k * DD + n] = a * W[k * DD + n];
  __syncthreads();
  float acc = 0.f;
  for (int j = 0; j < DD; ++j) acc += csh[j] * W[j * DD + k];
  cW[k] = acc;
}

// C[M,128] = A[M,128] @ B[128,128] + biasRow[128]
// One wave computes one 16x16 output tile via 32x V_WMMA_F32_16X16X4_F32.
__global__ void gemm_wmma(const float* __restrict__ A, const float* __restrict__ B,
                          const float* __restrict__ biasRow, float* __restrict__ C,
                          int mtiles) {
  int wid  = (blockIdx.x * blockDim.x + threadIdx.x) >> 5;
  int lane = threadIdx.x & 31;
  int mt = wid >> 3;       // 8 N-tiles (128/16)
  int nt = wid & 7;
  if (mt >= mtiles) return;  // whole-wave uniform; live waves keep EXEC all-1s
  int half = lane >> 4, lp = lane & 15;
  int m0 = mt * 16, n0 = nt * 16;

  float bval = biasRow[n0 + lp];
  v8f acc = { bval, bval, bval, bval, bval, bval, bval, bval };

  const float* arow = A + (long)(m0 + lp) * DD;
  const float* bcol = B + n0 + lp;
  #pragma unroll
  for (int k0 = 0; k0 < DD; k0 += 4) {
    int ka = k0 + 2 * half;
    // A 16x4 f32: lanes0-15 hold K=k0,k0+1 ; lanes16-31 hold K=k0+2,k0+3 (rows M=lp)
    v2f av; av.x = arow[ka]; av.y = arow[ka + 1];
    // B 4x16 f32: same K split, columns N=lp per lane
    v2f bv; bv.x = bcol[(long)ka * DD]; bv.y = bcol[(long)(ka + 1) * DD];
    acc = __builtin_amdgcn_wmma_f32_16x16x4_f32(false, av, false, bv,
                                                (short)0, acc, false, false);
  }
  // D layout: VGPR g -> row m0 + g + 8*half, col n0 + lp
  long base = (long)(m0 + 8 * half) * DD + n0 + lp;
  #pragma unroll
  for (int g = 0; g < 8; ++g) C[base + (long)g * DD] = acc[g];
}

// beta[he][h] = dot(eh[he, h*32 : h*32+32], att[h, 32:64])
__global__ void beta_kernel(const float* __restrict__ eh, const float* __restrict__ att,
                            float* __restrict__ beta) {
  int tid = blockIdx.x * blockDim.x + threadIdx.x;
  if (tid >= NHE * 4) return;
  int he = tid >> 2, h = tid & 3;
  const float4* er = (const float4*)(eh + (long)he * DD + h * 32);
  const float4* ar = (const float4*)(att + h * 64 + 32);
  float d = 0.f;
  #pragma unroll
  for (int i = 0; i < 8; ++i) {
    float4 xv = er[i], av = ar[i];
    d += xv.x * av.x + xv.y * av.y + xv.z * av.z + xv.w * av.w;
  }
  beta[tid] = d;
}

// Pass A: logits + leaky relu + segment max (ordered-uint atomicMax)
__global__ void edge_logits(const int* __restrict__ eidx, const float* __restrict__ xh,
                            const float* __restrict__ beta, const float* __restrict__ att,
                            float* __restrict__ alphaRaw, unsigned* __restrict__ mbuf) {
  int tid = blockIdx.x * blockDim.x + threadIdx.x;
  if (tid >= NE * 4) return;
  int e = tid >> 2, h = tid & 3;
  int src = eidx[e];
  int dst = eidx[NE + e];
  const float4* xr = (const float4*)(xh + (long)src * DD + h * 32);
  const float4* ar = (const float4*)(att + h * 64);
  float d = 0.f;
  #pragma unroll
  for (int i = 0; i < 8; ++i) {
    float4 xv = xr[i], av = ar[i];
    d += xv.x * av.x + xv.y * av.y + xv.z * av.z + xv.w * av.w;
  }
  d += beta[dst * 4 + h];
  d = (d >= 0.f) ? d : 0.2f * d;  // leaky relu
  alphaRaw[tid] = d;
  atomicMax(&mbuf[dst * 4 + h], encOrd(d));
}

// Pass B: exp(alpha - max) + segment sum
__global__ void edge_exp(const int* __restrict__ eidx, const float* __restrict__ alphaRaw,
                         const unsigned* __restrict__ mbuf, float* __restrict__ exBuf,
                         float* __restrict__ sbuf) {
  int tid = blockIdx.x * blockDim.x + threadIdx.x;
  if (tid >= NE * 4) return;
  int e = tid >> 2, h = tid & 3;
  int dst = eidx[NE + e];
  float ex = __expf(alphaRaw[tid] - decOrd(mbuf[dst * 4 + h]));
  exBuf[tid] = ex;
  atomicAdd(&sbuf[dst * 4 + h], ex);
}

// Pass C: acc[128] += sum_e softmax_weight(e,h) * xh[src(e), h*32+o]
// thread t <-> feature (h=t/32, o=t%32); register accumulate over an edge chunk.
__global__ void edge_accum(const int* __restrict__ eidx, const float* __restrict__ exBuf,
                           const float* __restrict__ sbuf, const float* __restrict__ xh,
                           float* __restrict__ acc) {
  int t = threadIdx.x;  // 128
  int h = t >> 5;
  long e0 = (long)blockIdx.x * CCHUNK;
  float r = 0.f;
  for (int i = 0; i < CCHUNK; ++i) {
    long e = e0 + i;
    if (e >= NE) break;
    int src = eidx[e];
    int dst = eidx[NE + e];
    float ex = exBuf[e * 4 + h];
    float w  = ex / (sbuf[dst * 4 + h] + 1e-16f);
    r += w * xh[(long)src * DD + t];
  }
  atomicAdd(&acc[t], r);
}

__global__ void finalize_kernel(const float* __restrict__ acc0, const float* __restrict__ acc1,
                                const float* __restrict__ cs0, const float* __restrict__ cs1,
                                const float* __restrict__ b0, const float* __restrict__ b1,
                                float* __restrict__ out) {
  int t = blockIdx.x * blockDim.x + threadIdx.x;
  if (t >= 512) return;
  const float invHE = 1.f / (float)NHE;
  const float invN  = 1.f / (float)NNODES;
  if (t < 128)      out[t] = acc0[t] * invHE + b0[t];
  else if (t < 256) out[t] = acc1[t - 128] * invHE + b1[t - 128];
  else if (t < 384) out[t] = cs0[t - 256] * invN;
  else              out[t] = cs1[t - 384] * invN;
}

extern "C" void kernel_launch(void* const* d_in, const int* in_sizes, int n_in,
                              void* d_out, int out_size, void* d_ws, size_t ws_size,
                              hipStream_t stream) {
  const float* x[2]    = { (const float*)d_in[0],  (const float*)d_in[2]  };
  const float* ea[2]   = { (const float*)d_in[1],  (const float*)d_in[3]  };
  const int*   ei[2]   = { (const int*)d_in[4],    (const int*)d_in[5]    };
  const float* gw[2]   = { (const float*)d_in[6],  (const float*)d_in[12] };
  const float* gb[2]   = { (const float*)d_in[7],  (const float*)d_in[13] };
  const float* gms[2]  = { (const float*)d_in[8],  (const float*)d_in[14] };
  const float* W[2]    = { (const float*)d_in[9],  (const float*)d_in[15] };
  const float* att[2]  = { (const float*)d_in[10], (const float*)d_in[16] };
  const float* bias[2] = { (const float*)d_in[11], (const float*)d_in[17] };
  float* out = (float*)d_out;

  // workspace layout (floats); big buffers reused across the two graphs
  float* ws   = (float*)d_ws;
  float* accB = ws;                 //   0: 2*128 message accumulators (persist)
  float* csB  = ws + 256;           // 256: 2*128 column sums (persist)
  float* csq  = ws + 512;           // 128 (reused)
  float* cW   = ws + 640;           // 128 (reused)
  float* z128 = ws + 768;           // 128 zeros
  float* Wsc  = ws + 896;           // 16384 (reused)
  float* beta = ws + 17280;         // 80000 (reused)
  unsigned* mb = (unsigned*)(ws + 97280);  // 80000 (reused)
  float* sb   = ws + 177280;        // 80000 (reused)
  float* aRaw = ws + 257280;        // 3,200,000 (reused)
  float* exB  = ws + 3457280;       // 3,200,000 (reused)
  float* eh   = ws + 6657280;       // 2,560,000 (reused)
  float* xh   = ws + 9217280;       // 12,800,000 (reused)  -> total ~84 MiB

  fill_f32<<<1, 128, 0, stream>>>(z128, 128, 0.f);

  for (int g = 0; g < 2; ++g) {
    float* acc = accB + g * 128;
    float* cs  = csB + g * 128;
    fill_f32<<<1, 128, 0, stream>>>(acc, 128, 0.f);
    fill_f32<<<1, 128, 0, stream>>>(cs, 128, 0.f);
    fill_f32<<<1, 128, 0, stream>>>(csq, 128, 0.f);
    fill_f32<<<(80000 + 255) / 256, 256, 0, stream>>>(sb, 80000, 0.f);
    fill_u32<<<(80000 + 255) / 256, 256, 0, stream>>>(mb, 80000, 0x007FFFFFu); // enc(-inf)

    colsum_kernel<<<512, 128, 0, stream>>>(x[g], NNODES, cs, csq);
    prep_kernel<<<1, 128, 0, stream>>>(cs, csq, gw[g], gb[g], gms[g], W[g],
                                       Wsc, cW, 1.f / (float)NNODES);

    // xh = gn(x) @ W  (folded: x @ Wsc + cW) ; eh = eattr @ W
    gemm_wmma<<<NNODES / 16, 256, 0, stream>>>(x[g], Wsc, cW, xh, NNODES / 16);
    gemm_wmma<<<NHE / 16, 256, 0, stream>>>(ea[g], W[g], z128, eh, NHE / 16);

    beta_kernel<<<(NHE * 4 + 255) / 256, 256, 0, stream>>>(eh, att[g], beta);
    edge_logits<<<(NE * 4 + 255) / 256, 256, 0, stream>>>(ei[g], xh, beta, att[g], aRaw, mb);
    edge_exp<<<(NE * 4 + 255) / 256, 256, 0, stream>>>(ei[g], aRaw, mb, exB, sb);
    edge_accum<<<(NE + CCHUNK - 1) / CCHUNK, 128, 0, stream>>>(ei[g], exB, sb, xh, acc);
  }

  finalize_kernel<<<1, 512, 0, stream>>>(accB, accB + 128, csB, csB + 128,
                                         bias[0], bias[1], out);
}